// SpatialDropoutX_36215164240425
// MI455X (gfx1250) — compile-verified
//
#include <hip/hip_runtime.h>
#include <hip/hip_bf16.h>
#include <stdint.h>

// out[b,c,t] = mask[b,c] * X[b,c,t], mask from distance of loc[c] to loc[drop_ids[b]].
// Bandwidth-bound: ~570MB traffic -> ~24.5us at 23.3 TB/s. Stream with B128 + NT,
// feed loads through gfx1250 async global->LDS pipeline (ASYNCcnt), per-wave
// double-buffered, no block barriers (each wave consumes only its own LDS slice).

typedef __attribute__((ext_vector_type(4))) float v4f;
typedef int v4i __attribute__((vector_size(16)));                 // matches builtin param
typedef __attribute__((address_space(1))) v4i gv4i;               // global int4
typedef __attribute__((address_space(3))) v4i lv4i;               // LDS int4

#define CHUNK_ELEMS 1024          // 256 lanes * 4 floats = 4KB per chunk
#define CHUNKS_PER_BLOCK 4        // 16KB of the row per block

#if __has_builtin(__builtin_amdgcn_global_load_async_to_lds_b128)
#define HAVE_ASYNC_LOAD 1
#else
#define HAVE_ASYNC_LOAD 0
#endif

#if __has_builtin(__builtin_amdgcn_s_wait_asynccnt)
#define HAVE_WAIT_ASYNC 1
#else
#define HAVE_WAIT_ASYNC 0
#endif

template <int N>
__device__ __forceinline__ void wait_asynccnt() {
#if HAVE_WAIT_ASYNC
  __builtin_amdgcn_s_wait_asynccnt(N);
#else
  asm volatile("s_wait_asynccnt %0" ::"n"(N));
#endif
}

#if HAVE_ASYNC_LOAD
__device__ __forceinline__ void async_cp_b128(const float* g, lv4i* l) {
  // global (AS1) -> LDS (AS3), 16 bytes per enabled lane, tracked by ASYNCcnt.
  // cpol=1: TH_NT -- X is a read-once 285MB stream, keep it from churning L2.
  __builtin_amdgcn_global_load_async_to_lds_b128(
      (gv4i*)(uintptr_t)g, l, /*imm offset*/ 0, /*cpol*/ 1);
}
#endif

__global__ __launch_bounds__(256) void SpatialDropoutX_kernel(
    const float* __restrict__ X, const float* __restrict__ loc,
    const int* __restrict__ drop_ids, float* __restrict__ out,
    int C, int T) {
  // 3D grid: x = row segment, y = channel c, z = batch b  (no integer division)
  const int c = blockIdx.y;
  const int b = blockIdx.z;
  const int r = b * C + c;

  // per-block mask (scalar-uniform work, ~6 flops; loc/drop_ids hit SMEM path)
  const int id = drop_ids[b];
  const float dx = loc[2 * c + 0] - loc[2 * id + 0];
  const float dy = loc[2 * c + 1] - loc[2 * id + 1];
  const float m = (sqrtf(dx * dx + dy * dy) >= 0.2f) ? 1.0f : 0.0f;

  const int t0 = blockIdx.x * (CHUNKS_PER_BLOCK * CHUNK_ELEMS);
  if (t0 >= T) return;
  int nch = (T - t0 + CHUNK_ELEMS - 1) / CHUNK_ELEMS;
  if (nch > CHUNKS_PER_BLOCK) nch = CHUNKS_PER_BLOCK;

  const long long rowBase = (long long)r * (long long)T;
  const float* gX = X + rowBase;
  float* gO = out + rowBase;
  const int tid = threadIdx.x;

#if HAVE_ASYNC_LOAD
  __shared__ v4f lds[2][256];  // 2 x 4KB double buffer; each wave owns its 512B slice
  lv4i* lds3 = (lv4i*)(uint32_t)(uintptr_t)&lds[0][0];

  // prime the pipeline
  {
    const int e = t0 + (tid << 2);
    if (e + 3 < T) async_cp_b128(gX + e, lds3 + tid);
  }
  for (int k = 0; k < nch; ++k) {
    if (k + 1 < nch) {
      const int e = t0 + ((k + 1) << 10) + (tid << 2);
      if (e + 3 < T) async_cp_b128(gX + e, lds3 + (((k + 1) & 1) << 8) + tid);
      wait_asynccnt<1>();  // in-order completion: chunk k has landed
    } else {
      wait_asynccnt<0>();
    }
    asm volatile("" ::: "memory");  // don't hoist LDS reads above the wait
    const int e = t0 + (k << 10) + (tid << 2);
    if (e + 3 < T) {
      v4f x = lds[k & 1][tid];
      x *= m;
      __builtin_nontemporal_store(x, (v4f*)(gO + e));
    } else if (e < T) {  // ragged tail (not hit for T=8192): direct scalar path
      for (int j = 0; j < 4 && e + j < T; ++j) {
        float xv = __builtin_nontemporal_load(gX + e + j);
        __builtin_nontemporal_store(xv * m, gO + e + j);
      }
    }
  }
#else
  // Fallback: direct B128 non-temporal stream (still bandwidth-optimal)
  for (int k = 0; k < nch; ++k) {
    const int e = t0 + (k << 10) + (tid << 2);
    if (e + 3 < T) {
      v4f x = __builtin_nontemporal_load((const v4f*)(gX + e));
      x *= m;
      __builtin_nontemporal_store(x, (v4f*)(gO + e));
    } else if (e < T) {
      for (int j = 0; j < 4 && e + j < T; ++j) {
        float xv = __builtin_nontemporal_load(gX + e + j);
        __builtin_nontemporal_store(xv * m, gO + e + j);
      }
    }
  }
#endif
}

extern "C" void kernel_launch(void* const* d_in, const int* in_sizes, int n_in,
                              void* d_out, int out_size, void* d_ws, size_t ws_size,
                              hipStream_t stream) {
  const float* X = (const float*)d_in[0];
  const float* loc = (const float*)d_in[1];
  const int* drop_ids = (const int*)d_in[2];
  float* out = (float*)d_out;

  const int C = in_sizes[1] / 2;   // loc is [C,2]
  const int B = in_sizes[2];       // drop_ids is [B]
  const long long nX = (long long)in_sizes[0];
  const int T = (int)(nX / ((long long)B * (long long)C));

  const int elemsPerBlock = CHUNKS_PER_BLOCK * CHUNK_ELEMS;  // 4096
  const int gx = (T + elemsPerBlock - 1) / elemsPerBlock;

  dim3 grid(gx, C, B);  // b, c straight from blockIdx -> no division in kernel
  SpatialDropoutX_kernel<<<grid, 256, 0, stream>>>(X, loc, drop_ids, out, C, T);
}